// GCNLayerWithSkip_84301618086372
// MI455X (gfx1250) — compile-verified
//
#include <hip/hip_runtime.h>
#include <math.h>

typedef float v2f __attribute__((ext_vector_type(2)));
typedef float v8f __attribute__((ext_vector_type(8)));

#define DD   96
#define STR  100           // padded LDS row stride (floats) for transposed W: conflict-free b64 loads
#define TILE 16
#define CT   (DD / TILE)   // 6 column tiles

// ---------------------------------------------------------------------------
// 1) zero the output accumulator, init degree = 1 (self-loop contribution)
// ---------------------------------------------------------------------------
__global__ void init_kernel(float* __restrict__ deg, float* __restrict__ out, int n) {
    int i = blockIdx.x * blockDim.x + threadIdx.x;
    if (i < n * DD) out[i] = 0.0f;
    if (i < n)      deg[i] = 1.0f;
}

// ---------------------------------------------------------------------------
// 2) degree count over destination nodes
// ---------------------------------------------------------------------------
__global__ void deg_kernel(const long long* __restrict__ dst, float* __restrict__ deg, int e) {
    int i = blockIdx.x * blockDim.x + threadIdx.x;
    if (i < e) atomicAdd(&deg[(int)dst[i]], 1.0f);
}

__global__ void rsqrt_kernel(float* __restrict__ deg, int n) {
    int i = blockIdx.x * blockDim.x + threadIdx.x;
    if (i < n) deg[i] = rsqrtf(deg[i]);   // deg >= 1 always (self-loop)
}

// ---------------------------------------------------------------------------
// Stage W transposed into LDS: sWt[col * STR + k] = W[k * DD + col].
// B operand for V_WMMA_F32_16X16X4_F32 then becomes one contiguous ds_load_b64.
// ---------------------------------------------------------------------------
__device__ __forceinline__ void stage_wt(float* sWt, const float* __restrict__ W,
                                         int tid, int nthreads) {
    for (int i = tid; i < DD * DD; i += nthreads) {
        int k = i / DD, c = i % DD;
        sWt[c * STR + k] = W[i];
    }
}

// ---------------------------------------------------------------------------
// 3) h = x @ W_gcn  via V_WMMA_F32_16X16X4_F32, one 16x16 tile per wave.
// ---------------------------------------------------------------------------
__global__ void gemm_h_kernel(const float* __restrict__ x, const float* __restrict__ W,
                              float* __restrict__ h, int n) {
    __shared__ float sWt[DD * STR];
    stage_wt(sWt, W, threadIdx.x, blockDim.x);
    __syncthreads();

    int wave = threadIdx.x >> 5;
    int lane = threadIdx.x & 31;
    int tile = blockIdx.x * (blockDim.x >> 5) + wave;
    int totalTiles = (n / TILE) * CT;
    if (tile >= totalTiles) return;           // wave-uniform guard, EXEC stays all-ones

    int rt = tile / CT, ct = tile % CT;
    int row0 = rt * TILE, col0 = ct * TILE;

    int ml = lane & 15;                        // row within tile (A) / col within tile (B)
    int kb = (lane >> 4) << 1;                 // lanes 16-31 hold K+2 / K+3
    const float* xrow = x + (size_t)(row0 + ml) * DD;
    const float* wcol = sWt + (size_t)(col0 + ml) * STR;

    v8f c = {};
    #pragma unroll
    for (int k = 0; k < DD; k += 4) {
        v2f a = *(const v2f*)&xrow[k + kb];     // global_load_b64
        v2f b = *(const v2f*)&wcol[k + kb];     // ds_load_b64 (contiguous, conflict-free)
        c = __builtin_amdgcn_wmma_f32_16x16x4_f32(false, a, false, b, (short)0, c, false, false);
    }

    int mBase = row0 + ((lane >> 4) << 3);     // C/D layout: lanes 16-31 hold rows M+8..M+15
    int nOut  = col0 + ml;
    #pragma unroll
    for (int i = 0; i < 8; ++i)
        h[(size_t)(mBase + i) * DD + nOut] = c[i];
}

// ---------------------------------------------------------------------------
// 4) edge scatter: one wave per edge, lanes cover the 96 columns (3 x 32),
//    coalesced gather of h[src] and coalesced f32 atomics into out[dst].
// ---------------------------------------------------------------------------
__global__ void scatter_kernel(const long long* __restrict__ src, const long long* __restrict__ dst,
                               const float* __restrict__ h, const float* __restrict__ dinv,
                               float* __restrict__ out, int e) {
    int t    = blockIdx.x * blockDim.x + threadIdx.x;
    int edge = t >> 5;
    int lane = t & 31;
    if (edge >= e) return;
    int s = (int)src[edge];
    int d = (int)dst[edge];
    float w = dinv[s] * dinv[d];
    const float* hp = h + (size_t)s * DD;
    float*       op = out + (size_t)d * DD;
    #pragma unroll
    for (int j = 0; j < 3; ++j) {
        int cidx = lane + j * 32;
        atomicAdd(&op[cidx], hp[cidx] * w);
    }
}

// ---------------------------------------------------------------------------
// 5) fused epilogue: skip = x @ W_lin (WMMA) ; out = mish(agg + selfloop + biases + skip)
// ---------------------------------------------------------------------------
__global__ void final_kernel(const float* __restrict__ x, const float* __restrict__ W,
                             const float* __restrict__ bg, const float* __restrict__ bl,
                             const float* __restrict__ h, const float* __restrict__ dinv,
                             float* __restrict__ out, int n) {
    __shared__ float sWt[DD * STR];
    stage_wt(sWt, W, threadIdx.x, blockDim.x);
    __syncthreads();

    int wave = threadIdx.x >> 5;
    int lane = threadIdx.x & 31;
    int tile = blockIdx.x * (blockDim.x >> 5) + wave;
    int totalTiles = (n / TILE) * CT;
    if (tile >= totalTiles) return;

    int rt = tile / CT, ct = tile % CT;
    int row0 = rt * TILE, col0 = ct * TILE;

    int ml = lane & 15;
    int kb = (lane >> 4) << 1;
    const float* xrow = x + (size_t)(row0 + ml) * DD;
    const float* wcol = sWt + (size_t)(col0 + ml) * STR;

    v8f c = {};
    #pragma unroll
    for (int k = 0; k < DD; k += 4) {
        v2f a = *(const v2f*)&xrow[k + kb];
        v2f b = *(const v2f*)&wcol[k + kb];
        c = __builtin_amdgcn_wmma_f32_16x16x4_f32(false, a, false, b, (short)0, c, false, false);
    }

    int mBase = row0 + ((lane >> 4) << 3);
    int nOut  = col0 + ml;
    float bsum = bg[nOut] + bl[nOut];
    #pragma unroll
    for (int i = 0; i < 8; ++i) {
        int m = mBase + i;
        float dv = dinv[m];
        size_t idx = (size_t)m * DD + nOut;
        float y = out[idx]                    // scattered neighbor aggregate
                + h[idx] * dv * dv            // self-loop term (src==dst, norm=dinv^2)
                + c[i]                        // skip GEMM
                + bsum;                       // b_gcn + b_lin
        float sp = (y > 20.0f) ? y : log1pf(__expf(y));   // softplus (stable)
        out[idx] = y * tanhf(sp);                          // mish
    }
}

// ---------------------------------------------------------------------------
extern "C" void kernel_launch(void* const* d_in, const int* in_sizes, int n_in,
                              void* d_out, int out_size, void* d_ws, size_t ws_size,
                              hipStream_t stream) {
    const float*     x   = (const float*)d_in[0];
    const long long* ei  = (const long long*)d_in[1];
    const float*     Wg  = (const float*)d_in[2];
    const float*     bg  = (const float*)d_in[3];
    const float*     Wl  = (const float*)d_in[4];
    const float*     bl  = (const float*)d_in[5];
    float*           out = (float*)d_out;

    int n = in_sizes[0] / DD;      // 50000
    int e = in_sizes[1] / 2;       // 800000
    const long long* src = ei;
    const long long* dst = ei + e;

    // workspace: [ deg/dinv : n floats (rounded) ][ h : n*96 floats ]
    float* deg = (float*)d_ws;
    float* h   = deg + (((size_t)n + 63) & ~(size_t)63);

    init_kernel <<<(n * DD + 255) / 256, 256, 0, stream>>>(deg, out, n);
    deg_kernel  <<<(e + 255) / 256,      256, 0, stream>>>(dst, deg, e);
    rsqrt_kernel<<<(n + 255) / 256,      256, 0, stream>>>(deg, n);

    int tiles  = (n / TILE) * CT;           // 3125 * 6 = 18750 (n divisible by 16)
    int blocks = (tiles + 7) / 8;           // 8 waves (tiles) per 256-thread block
    gemm_h_kernel<<<blocks, 256, 0, stream>>>(x, Wg, h, n);

    long long sthreads = (long long)e * 32; // one wave per edge
    scatter_kernel<<<(int)((sthreads + 255) / 256), 256, 0, stream>>>(src, dst, h, deg, out, e);

    final_kernel<<<blocks, 256, 0, stream>>>(x, Wl, bg, bl, h, deg, out, n);
}